// CustomWord2Vec_51204600103374
// MI455X (gfx1250) — compile-verified
//
#include <hip/hip_runtime.h>
#include <hip/hip_bf16.h>

// CDNA5 / gfx1250: wave32, WMMA 16x16x4 F32.
typedef __attribute__((ext_vector_type(2))) float v2f;
typedef __attribute__((ext_vector_type(8))) float v8f;

#define EMB 128

// One wave of 32 lanes computes sigmoid(dot(wv_i, cv_i)) for 16 batch pairs.
// A-matrix (16x4 slice of word rows) and B-matrix (4x16 slice of context rows,
// B[k][j] = cv_j[k]) are fed to V_WMMA_F32_16X16X4_F32; the diagonal of the
// 16x16 accumulator D holds the 16 dot products after K=128 in 32 steps.
__global__ __launch_bounds__(256)
void w2v_dot_wmma(const int* __restrict__ widx,
                  const int* __restrict__ cidx,
                  const float* __restrict__ table,
                  float* __restrict__ out,
                  int batch) {
  const int lane = threadIdx.x & 31;
  const int wave = (blockIdx.x * blockDim.x + threadIdx.x) >> 5;
  const int base = wave << 4;                 // 16 pairs per wave
  if (base + 16 > batch) return;              // wave-uniform: EXEC stays all-1s

  const int row = lane & 15;                  // which pair this lane serves
  const int koff = (lane >> 4) << 1;          // lanes 0-15: K+0/1, lanes 16-31: K+2/3

  const long wr = (long)widx[base + row];
  const long cr = (long)cidx[base + row];
  const float* wp = table + wr * (long)EMB + koff;
  const float* cp = table + cr * (long)EMB + koff;

  v8f acc = {0.f, 0.f, 0.f, 0.f, 0.f, 0.f, 0.f, 0.f};

#pragma unroll
  for (int k = 0; k < EMB; k += 4) {
    // ISA 32-bit A 16x4 layout: VGPR0 = K(+0 / +2), VGPR1 = K(+1 / +3)
    // per half-wave; one global_load_b64 per operand per chunk.
    v2f a = *(const v2f*)(wp + k);
    v2f b = *(const v2f*)(cp + k);
    // 8 args: (neg_a, A, neg_b, B, c_mod, C, reuse_a, reuse_b)
    acc = __builtin_amdgcn_wmma_f32_16x16x4_f32(
        false, a, false, b, (short)0, acc, false, false);
  }

  // Diagonal extraction per documented 16x16 f32 C/D layout:
  //   lanes 0-7  hold D[i,i], i = lane,     in acc[lane]
  //   lanes 24-31 hold D[i,i], i = lane-16, in acc[lane-24]  (== acc[lane&7])
  const int r8 = lane & 7;
  float d = acc[0];
  d = (r8 == 1) ? acc[1] : d;
  d = (r8 == 2) ? acc[2] : d;
  d = (r8 == 3) ? acc[3] : d;
  d = (r8 == 4) ? acc[4] : d;
  d = (r8 == 5) ? acc[5] : d;
  d = (r8 == 6) ? acc[6] : d;
  d = (r8 == 7) ? acc[7] : d;

  const float s = 1.0f / (1.0f + __expf(-d));

  if (lane < 8) {
    out[base + lane] = s;
  } else if (lane >= 24) {
    out[base + lane - 16] = s;
  }
}

extern "C" void kernel_launch(void* const* d_in, const int* in_sizes, int n_in,
                              void* d_out, int out_size, void* d_ws, size_t ws_size,
                              hipStream_t stream) {
  (void)n_in; (void)d_ws; (void)ws_size; (void)out_size;
  const int*   widx  = (const int*)d_in[0];    // word_vector  (BATCH int32)
  const int*   cidx  = (const int*)d_in[1];    // context_vector (BATCH int32)
  const float* table = (const float*)d_in[2];  // lookup_weight (VOCAB x 128 f32)
  float*       out   = (float*)d_out;          // BATCH f32

  const int batch = in_sizes[0];
  // 256 threads = 8 wave32 = 128 pairs per block.
  const int pairs_per_block = (256 / 32) * 16;
  const int grid = (batch + pairs_per_block - 1) / pairs_per_block;
  w2v_dot_wmma<<<grid, 256, 0, stream>>>(widx, cidx, table, out, batch);
}